// GCN_38517266710932
// MI455X (gfx1250) — compile-verified
//
#include <hip/hip_runtime.h>
#include <hip/hip_bf16.h>

typedef __attribute__((ext_vector_type(2))) float v2f;
typedef __attribute__((ext_vector_type(8))) float v8f;

#define IN_CH  128
#define HID_CH 128
#define OUT_CH 64

// ---------------------------------------------------------------------------
// Degree / normalization:  dinv[i] = rsqrt(1 + indegree(i))
// ---------------------------------------------------------------------------
__global__ void deg_init_kernel(float* __restrict__ dinv, int n) {
    int i = blockIdx.x * blockDim.x + threadIdx.x;
    if (i < n) dinv[i] = 1.0f;  // self-loop contribution
}

__global__ void deg_count_kernel(const int* __restrict__ dst, float* __restrict__ dinv, int nE) {
    int i = blockIdx.x * blockDim.x + threadIdx.x;
    if (i < nE) unsafeAtomicAdd(&dinv[dst[i]], 1.0f);
}

__global__ void deg_rsqrt_kernel(float* __restrict__ dinv, int n) {
    int i = blockIdx.x * blockDim.x + threadIdx.x;
    if (i < n) dinv[i] = rsqrtf(dinv[i]);  // deg >= 1 always
}

// ---------------------------------------------------------------------------
// FP32 WMMA GEMM:  C[M,N] = A[M,K] @ B[K,N]
// K, N compile-time -> fully unrolled: 32 v_wmma_f32_16x16x4_f32 per wave,
// immediate-offset B loads, 32-bit addressing.
// One wave computes one 16x16 tile; 4 waves/block share the same A rows.
// ---------------------------------------------------------------------------
template <int K, int N>
__global__ __launch_bounds__(128)
void gemm_f32_wmma_kernel(const float* __restrict__ A,
                          const float* __restrict__ B,
                          float* __restrict__ C) {
    const int lane  = threadIdx.x & 31;
    const int wave  = threadIdx.x >> 5;
    const int mBase = blockIdx.x << 4;
    const int nBase = ((blockIdx.y << 2) + wave) << 4;
    const int l15   = lane & 15;
    const int kh    = (lane >> 4) << 1;   // 0 for lanes 0-15, 2 for lanes 16-31

    // A fragment: lane holds row (mBase+l15), K-pair starting at kh
    const float* aRow = A + (mBase + l15) * K + kh;
    // B fragment: lane holds col (nBase+l15), K-pair rows starting at kh
    const float* bCol = B + kh * N + nBase + l15;

    v8f acc = {};
#pragma unroll
    for (int k = 0; k < K; k += 4) {
        v2f a = *(const v2f*)(aRow + k);          // global_load_b64 (8B aligned)
        v2f b;
        b.x = bCol[k * N];                        // immediate-offset b32 loads
        b.y = bCol[k * N + N];
        acc = __builtin_amdgcn_wmma_f32_16x16x4_f32(
            /*neg_a=*/false, a, /*neg_b=*/false, b,
            /*c_mod=*/(short)0, acc, /*reuse_a=*/false, /*reuse_b=*/false);
    }

    // C/D layout: VGPR r -> row r (lanes 0-15) / row r+8 (lanes 16-31)
    float* cPtr = C + (mBase + ((lane >> 4) << 3)) * N + nBase + l15;
#pragma unroll
    for (int r = 0; r < 8; ++r) cPtr[r * N] = acc[r];
}

// ---------------------------------------------------------------------------
// out[i,:] = xw[i,:] * dinv[i]^2   (self-loop message; also zero-inits out)
// Vectorized float4; Cq = channels/4.
// ---------------------------------------------------------------------------
__global__ void selfloop_init_kernel(const float4* __restrict__ xw,
                                     const float* __restrict__ dinv,
                                     float4* __restrict__ out,
                                     int total4, int Cq) {
    int i = blockIdx.x * blockDim.x + threadIdx.x;
    if (i >= total4) return;
    float d = dinv[i / Cq];
    float s = d * d;
    float4 v = xw[i];
    v.x *= s; v.y *= s; v.z *= s; v.w *= s;
    out[i] = v;
}

// ---------------------------------------------------------------------------
// Edge scatter-add: one wave per edge, CPL channels per lane (C = 32*CPL).
// out[dst] += xw[src] * dinv[src]*dinv[dst]
// Gather is an explicit 128b/64b vector load (one cacheline per lane-group).
// ---------------------------------------------------------------------------
template <int CPL>
__global__ __launch_bounds__(256)
void scatter_edges_kernel(const float* __restrict__ xw,
                          const float* __restrict__ dinv,
                          const int* __restrict__ src,
                          const int* __restrict__ dst,
                          float* __restrict__ out,
                          int nE) {
    int gid  = blockIdx.x * blockDim.x + threadIdx.x;
    int e    = gid >> 5;
    int lane = gid & 31;
    if (e >= nE) return;
    int s = src[e];
    int d = dst[e];
    float norm = dinv[s] * dinv[d];
    const float* xp = xw  + s * (32 * CPL) + lane * CPL;
    float*       op = out + d * (32 * CPL) + lane * CPL;
    if (CPL == 4) {
        float4 m = *(const float4*)xp;            // global_load_b128
        unsafeAtomicAdd(&op[0], m.x * norm);
        unsafeAtomicAdd(&op[1], m.y * norm);
        unsafeAtomicAdd(&op[2], m.z * norm);
        unsafeAtomicAdd(&op[3], m.w * norm);
    } else {
        float2 m = *(const float2*)xp;            // global_load_b64
        unsafeAtomicAdd(&op[0], m.x * norm);
        unsafeAtomicAdd(&op[1], m.y * norm);
    }
}

// ---------------------------------------------------------------------------
// h = (relu?) (h + bias)   — C is a power of two, mask = C-1 (on float4 lanes)
// ---------------------------------------------------------------------------
template <bool RELU>
__global__ void bias_act_kernel(float4* __restrict__ h,
                                const float4* __restrict__ bias4,
                                int total4, int mask4) {
    int i = blockIdx.x * blockDim.x + threadIdx.x;
    if (i >= total4) return;
    float4 v = h[i];
    float4 b = bias4[i & mask4];
    v.x += b.x; v.y += b.y; v.z += b.z; v.w += b.w;
    if (RELU) {
        v.x = v.x > 0.0f ? v.x : 0.0f;
        v.y = v.y > 0.0f ? v.y : 0.0f;
        v.z = v.z > 0.0f ? v.z : 0.0f;
        v.w = v.w > 0.0f ? v.w : 0.0f;
    }
    h[i] = v;
}

// ---------------------------------------------------------------------------
extern "C" void kernel_launch(void* const* d_in, const int* in_sizes, int n_in,
                              void* d_out, int out_size, void* d_ws, size_t ws_size,
                              hipStream_t stream) {
    const float* x  = (const float*)d_in[0];
    const int*   ei = (const int*)d_in[1];
    const float* W1 = (const float*)d_in[2];
    const float* b1 = (const float*)d_in[3];
    const float* W2 = (const float*)d_in[4];
    const float* b2 = (const float*)d_in[5];
    float* out = (float*)d_out;

    const int nNodes = in_sizes[0] / IN_CH;   // 100000
    const int nEdges = in_sizes[1] / 2;       // 1000000
    const int* src = ei;
    const int* dst = ei + nEdges;

    // Workspace layout: dinv | bufA (xw1 / xw2) | bufB (h1)
    char* ws = (char*)d_ws;
    float* dinv = (float*)ws;
    size_t off1 = (((size_t)nNodes * sizeof(float)) + 255) & ~(size_t)255;
    float* bufA = (float*)(ws + off1);
    size_t off2 = off1 + (size_t)nNodes * HID_CH * sizeof(float);
    float* bufB = (float*)(ws + off2);

    const int T = 256;
    // --- degrees ---
    deg_init_kernel<<<(nNodes + T - 1) / T, T, 0, stream>>>(dinv, nNodes);
    deg_count_kernel<<<(nEdges + T - 1) / T, T, 0, stream>>>(dst, dinv, nEdges);
    deg_rsqrt_kernel<<<(nNodes + T - 1) / T, T, 0, stream>>>(dinv, nNodes);

    // --- layer 1: xw1 = x @ W1 (WMMA f32) ---
    dim3 g1(nNodes / 16, HID_CH / 64);  // 4 waves/block cover 64 cols
    gemm_f32_wmma_kernel<IN_CH, HID_CH><<<g1, 128, 0, stream>>>(x, W1, bufA);

    // h1 = self-loop + scatter-add + bias + relu
    int total1  = nNodes * HID_CH;
    int total1q = total1 / 4;
    selfloop_init_kernel<<<(total1q + T - 1) / T, T, 0, stream>>>(
        (const float4*)bufA, dinv, (float4*)bufB, total1q, HID_CH / 4);
    long long sthreads = (long long)nEdges * 32;
    scatter_edges_kernel<4><<<(int)((sthreads + T - 1) / T), T, 0, stream>>>(
        bufA, dinv, src, dst, bufB, nEdges);
    bias_act_kernel<true><<<(total1q + T - 1) / T, T, 0, stream>>>(
        (float4*)bufB, (const float4*)b1, total1q, HID_CH / 4 - 1);

    // --- layer 2: xw2 = h1 @ W2 (WMMA f32), reuse bufA ---
    dim3 g2(nNodes / 16, OUT_CH / 64);  // grid.y = 1, 4 waves cover 64 cols
    gemm_f32_wmma_kernel<HID_CH, OUT_CH><<<g2, 128, 0, stream>>>(bufB, W2, bufA);

    // z = self-loop + scatter-add + bias  (directly into d_out)
    int total2  = nNodes * OUT_CH;
    int total2q = total2 / 4;
    selfloop_init_kernel<<<(total2q + T - 1) / T, T, 0, stream>>>(
        (const float4*)bufA, dinv, (float4*)out, total2q, OUT_CH / 4);
    scatter_edges_kernel<2><<<(int)((sthreads + T - 1) / T), T, 0, stream>>>(
        bufA, dinv, src, dst, out, nEdges);
    bias_act_kernel<false><<<(total2q + T - 1) / T, T, 0, stream>>>(
        (float4*)out, (const float4*)b2, total2q, OUT_CH / 4 - 1);
}